// Unsupervised_49117245997509
// MI455X (gfx1250) — compile-verified
//
#include <hip/hip_runtime.h>

#define BB 8
#define HH 448
#define WW 1024
#define HWP (HH * WW)
#define RR 3
#define TX 64
#define TY 16
#define HX (TX + 6)
#define HY (TY + 6)
#define NHALO (HX * HY)

// -------- block-wide sum reduction (wave32 shuffles + 8-wave LDS stage) -----
__device__ __forceinline__ float blockReduceSum(float v, float* sred) {
#pragma unroll
  for (int off = 16; off > 0; off >>= 1) v += __shfl_xor(v, off, 32);
  const int lane = threadIdx.x & 31;
  const int wid = threadIdx.x >> 5;
  if (lane == 0) sred[wid] = v;
  __syncthreads();
  float r = 0.0f;
  if (wid == 0) {
    r = (lane < (int)(blockDim.x >> 5)) ? sred[lane] : 0.0f;
#pragma unroll
    for (int off = 4; off > 0; off >>= 1) r += __shfl_xor(r, off, 32);
  }
  __syncthreads();  // allow sred reuse by a subsequent reduction
  return r;         // valid in wave 0
}

__global__ void zero_acc_kernel(double* acc) {
  if (threadIdx.x < 4) acc[threadIdx.x] = 0.0;
}

// ---- fused kernel: census + smoothness + EPE, single pass over all inputs --
__global__ __launch_bounds__(256) void fused_loss_kernel(
    const float* __restrict__ flow, const float* __restrict__ tgt,
    const float* __restrict__ img1, const float* __restrict__ img2,
    double* __restrict__ acc) {
  __shared__ float sU[NHALO];   // flow u halo
  __shared__ float sV[NHALO];   // flow v halo
  __shared__ float sR[NHALO];   // img1 R halo
  __shared__ float sGr[NHALO];  // img1 G halo
  __shared__ float sB[NHALO];   // img1 B halo
  __shared__ float sG1[NHALO];  // gray(img1)*255 halo
  __shared__ float sG2[NHALO];  // gray(warp(img2))*255 halo
  __shared__ float sred[8];

  const int b = blockIdx.z;
  const int x0 = blockIdx.x * TX;
  const int y0 = blockIdx.y * TY;
  const int tid = threadIdx.x;

  // ---- stage flow(u,v) + img1(R,G,B) halos into LDS via async global->LDS
  const unsigned long long fbase =
      (unsigned long long)(flow + (size_t)b * 2 * HWP);  // u-plane base
  const unsigned long long ibase =
      (unsigned long long)(img1 + (size_t)b * 3 * HWP);  // R-plane base
  for (int i = tid; i < NHALO; i += 256) {
    const int hy = i / HX, hx = i - hy * HX;
    const int gy = min(max(y0 - RR + hy, 0), HH - 1);
    const int gx = min(max(x0 - RR + hx, 0), WW - 1);
    const int off0 = (gy * WW + gx) * 4;
    const int off1 = off0 + HWP * 4;
    const int off2 = off0 + 2 * HWP * 4;
    const unsigned int ldsU = (unsigned int)(size_t)(&sU[i]);
    const unsigned int ldsV = (unsigned int)(size_t)(&sV[i]);
    const unsigned int ldsR = (unsigned int)(size_t)(&sR[i]);
    const unsigned int ldsG = (unsigned int)(size_t)(&sGr[i]);
    const unsigned int ldsB = (unsigned int)(size_t)(&sB[i]);
    asm volatile("global_load_async_to_lds_b32 %0, %1, %2"
                 :: "v"(ldsU), "v"(off0), "s"(fbase) : "memory");
    asm volatile("global_load_async_to_lds_b32 %0, %1, %2"
                 :: "v"(ldsV), "v"(off1), "s"(fbase) : "memory");
    asm volatile("global_load_async_to_lds_b32 %0, %1, %2"
                 :: "v"(ldsR), "v"(off0), "s"(ibase) : "memory");
    asm volatile("global_load_async_to_lds_b32 %0, %1, %2"
                 :: "v"(ldsG), "v"(off1), "s"(ibase) : "memory");
    asm volatile("global_load_async_to_lds_b32 %0, %1, %2"
                 :: "v"(ldsB), "v"(off2), "s"(ibase) : "memory");
  }
  asm volatile("s_wait_asynccnt 0" ::: "memory");
  __syncthreads();

  // ---- per-halo-pixel: gray1 (LDS->LDS) and warped gray2 (img2 gathers) ----
  const float* __restrict__ i2 = img2 + (size_t)b * 3 * HWP;
  for (int i = tid; i < NHALO; i += 256) {
    const int hy = i / HX, hx = i - hy * HX;
    const int gy = min(max(y0 - RR + hy, 0), HH - 1);
    const int gx = min(max(x0 - RR + hx, 0), WW - 1);

    // grayscale of img1 (overlaps with the gather latency below)
    sG1[i] = (0.299f * sR[i] + 0.587f * sGr[i] + 0.114f * sB[i]) * 255.0f;

    // bilinear warp of img2 at (gy,gx) displaced by flow
    const float x = (float)gx + sU[i];
    const float y = (float)gy + sV[i];
    const float xf = floorf(x), yf = floorf(y);
    const float wx = x - xf, wy = y - yf;
    const int x0i = min(max((int)xf, 0), WW - 1);
    const int x1i = min(x0i + 1, WW - 1);
    const int y0i = min(max((int)yf, 0), HH - 1);
    const int y1i = min(y0i + 1, HH - 1);
    const int p00 = y0i * WW + x0i, p01 = y0i * WW + x1i;
    const int p10 = y1i * WW + x0i, p11 = y1i * WW + x1i;
    const float g00 = 0.299f * i2[p00] + 0.587f * i2[p00 + HWP] + 0.114f * i2[p00 + 2 * HWP];
    const float g01 = 0.299f * i2[p01] + 0.587f * i2[p01 + HWP] + 0.114f * i2[p01 + 2 * HWP];
    const float g10 = 0.299f * i2[p10] + 0.587f * i2[p10 + HWP] + 0.114f * i2[p10 + 2 * HWP];
    const float g11 = 0.299f * i2[p11] + 0.587f * i2[p11 + HWP] + 0.114f * i2[p11 + 2 * HWP];
    const float val = g00 * (1.0f - wx) * (1.0f - wy) + g01 * wx * (1.0f - wy) +
                      g10 * (1.0f - wx) * wy + g11 * wx * wy;
    sG2[i] = val * 255.0f;
  }
  __syncthreads();

  // ---- per inner pixel: census (masked) + smoothness + EPE ----------------
  const float* __restrict__ tb = tgt + (size_t)b * 2 * HWP;
  float cacc = 0.0f, sxacc = 0.0f, syacc = 0.0f, seacc = 0.0f;
  for (int k = 0; k < (TX * TY) / 256; ++k) {
    const int p = k * 256 + tid;
    const int iy = p >> 6;          // TX = 64
    const int ix = p & (TX - 1);
    const int gy = y0 + iy, gx = x0 + ix;
    const int gp = gy * WW + gx;
    const int hc = (iy + RR) * HX + (ix + RR);
    const float u = sU[hc], v = sV[hc];

    // EPE (only place target1 is ever read; coalesced)
    {
      const float du = u - tb[gp];
      const float dv = v - tb[gp + HWP];
      seacc += sqrtf(du * du + dv * dv);
    }

    // first-order edge-aware smoothness (x and y pairs, whole image)
    if (gx < WW - 1) {
      const float w = expf(-(fabsf(sR[hc] - sR[hc + 1]) + fabsf(sGr[hc] - sGr[hc + 1]) +
                             fabsf(sB[hc] - sB[hc + 1])) * (4.0f / 3.0f));
      const float fdu = u - sU[hc + 1];
      const float fdv = v - sV[hc + 1];
      sxacc += w * (powf(fdu * fdu + 1e-6f, 0.45f) + powf(fdv * fdv + 1e-6f, 0.45f));
    }
    if (gy < HH - 1) {
      const float w = expf(-(fabsf(sR[hc] - sR[hc + HX]) + fabsf(sGr[hc] - sGr[hc + HX]) +
                             fabsf(sB[hc] - sB[hc + HX])) * (4.0f / 3.0f));
      const float fdu = u - sU[hc + HX];
      const float fdv = v - sV[hc + HX];
      syacc += w * (powf(fdu * fdu + 1e-6f, 0.45f) + powf(fdv * fdv + 1e-6f, 0.45f));
    }

    // masked 7x7 soft census
    const float xw = (float)gx + u, yw = (float)gy + v;
    const bool inside = (xw >= 0.0f) && (xw <= (float)(WW - 1)) &&
                        (yw >= 0.0f) && (yw <= (float)(HH - 1));
    const bool inner = (gy >= RR) && (gy < HH - RR) && (gx >= RR) && (gx < WW - RR);
    if (inside && inner) {
      const float c1 = sG1[hc], c2 = sG2[hc];
      float dist = 0.0f;
#pragma unroll
      for (int dy = -RR; dy <= RR; ++dy) {
#pragma unroll
        for (int dx = -RR; dx <= RR; ++dx) {
          const int hi = hc + dy * HX + dx;
          const float d1 = sG1[hi] - c1;
          const float t1 = d1 * rsqrtf(0.81f + d1 * d1);
          const float d2 = sG2[hi] - c2;
          const float t2 = d2 * rsqrtf(0.81f + d2 * d2);
          float dd = t1 - t2;
          dd *= dd;
          dist += dd / (0.1f + dd);
        }
      }
      cacc += powf(dist * dist + 1e-6f, 0.45f);
    }
  }

  float t;
  t = blockReduceSum(cacc, sred);
  if (tid == 0) atomicAdd(acc + 0, (double)t);
  t = blockReduceSum(sxacc, sred);
  if (tid == 0) atomicAdd(acc + 1, (double)t);
  t = blockReduceSum(syacc, sred);
  if (tid == 0) atomicAdd(acc + 2, (double)t);
  t = blockReduceSum(seacc, sred);
  if (tid == 0) atomicAdd(acc + 3, (double)t);
}

// -------- finalize: apply reference denominators, emit (energy, epe) -------
__global__ void finalize_kernel(const double* __restrict__ acc,
                                float* __restrict__ out) {
  if (threadIdx.x == 0) {
    const double npix = (double)BB * (double)HWP;
    const double census = acc[0] / npix;
    const double sx = acc[1] / ((double)BB * 2.0 * (double)HH * (double)(WW - 1));
    const double sy = acc[2] / ((double)BB * 2.0 * (double)(HH - 1) * (double)WW);
    const double smooth = 0.5 * (sx + sy);
    out[0] = (float)(census + smooth);
    out[1] = (float)(acc[3] / npix);
  }
}

extern "C" void kernel_launch(void* const* d_in, const int* in_sizes, int n_in,
                              void* d_out, int out_size, void* d_ws, size_t ws_size,
                              hipStream_t stream) {
  const float* flow = (const float*)d_in[0];   // (8,2,448,1024)
  const float* tgt  = (const float*)d_in[1];   // (8,2,448,1024)
  const float* img1 = (const float*)d_in[2];   // (8,3,448,1024)
  const float* img2 = (const float*)d_in[3];   // (8,3,448,1024)
  float* out = (float*)d_out;                  // [energy, epe]
  double* acc = (double*)d_ws;                 // 4 double accumulators

  zero_acc_kernel<<<1, 64, 0, stream>>>(acc);

  dim3 g(WW / TX, HH / TY, BB);                // 16 x 28 x 8 tiles
  fused_loss_kernel<<<g, 256, 0, stream>>>(flow, tgt, img1, img2, acc);

  finalize_kernel<<<1, 32, 0, stream>>>(acc, out);
}